// NeuralExecutor3_17420387352973
// MI455X (gfx1250) — compile-verified
//
#include <hip/hip_runtime.h>
#include <math.h>

// Problem constants (match reference)
#define BB 8
#define NN 256
#define LL 32
#define MM 96      // message dim
// DIN = 98, W slices: [0:33]=z_i, [33:66]=z_j, [66:98]=enc_e

typedef float v2f __attribute__((ext_vector_type(2)));
typedef float v8f __attribute__((ext_vector_type(8)));

// fragment-layout strides (in floats)
#define W2G 224          // W2 group stride: 96*2 data + 32 pad -> %64 == 32 banks
#define RFG 32           // strip group stride: 16 rows * 2
#define NKK 24           // K chunks of 4 (K = 96)

__device__ __forceinline__ float neg_inf() { return -__builtin_inff(); }

// ---------------------------------------------------------------------------
// Kernel 1: per-node projections of z=[xs,hs] through both encoder branches.
// A0 = z@Wlin[0:33], B0 = z@Wlin[33:66], A1 = z@W1[0:33], B1 = z@W1[33:66]
// grid = B*N blocks, 96 threads
// ---------------------------------------------------------------------------
__global__ void ne3_pre(const float* __restrict__ xs, const float* __restrict__ hs,
                        const float* __restrict__ Wlin, const float* __restrict__ W1,
                        float* __restrict__ A0, float* __restrict__ B0,
                        float* __restrict__ A1, float* __restrict__ B1) {
  __shared__ float z[34];
  const int bi = blockIdx.x;
  const int tid = threadIdx.x;
  if (tid < 33) z[tid] = (tid == 0) ? xs[bi] : hs[bi * LL + tid - 1];
  __syncthreads();
  const int c = tid;  // 0..95
  float a0 = 0.f, b0 = 0.f, a1 = 0.f, b1 = 0.f;
  for (int k = 0; k < 33; ++k) {
    const float zk = z[k];
    a0 += zk * Wlin[k * MM + c];
    b0 += zk * Wlin[(33 + k) * MM + c];
    a1 += zk * W1[k * MM + c];
    b1 += zk * W1[(33 + k) * MM + c];
  }
  const int o = bi * MM + c;
  A0[o] = a0; B0[o] = b0; A1[o] = a1; B1[o] = b1;
}

// ---------------------------------------------------------------------------
// Kernel 1b: weight-only folded constants.
// c0 = W_e @ Wlin[66:98]  (96)   c1 = W_e @ W1[66:98] (96)
// cm = W_e @ Wtm[64:96]   (32)   wtt = Wtu @ Wterm    (32)   cp = W_e@Wpred[64:96]
// 1 block, 96 threads
// ---------------------------------------------------------------------------
__global__ void ne3_consts(const float* __restrict__ We, const float* __restrict__ Wlin,
                           const float* __restrict__ W1, const float* __restrict__ Wtm,
                           const float* __restrict__ Wtu, const float* __restrict__ Wterm,
                           const float* __restrict__ Wpred,
                           float* __restrict__ c0, float* __restrict__ c1,
                           float* __restrict__ cm, float* __restrict__ wtt,
                           float* __restrict__ cp) {
  const int c = threadIdx.x;  // 0..95
  float s0 = 0.f, s1 = 0.f;
  for (int h = 0; h < LL; ++h) {
    const float w = We[h];
    s0 += w * Wlin[(66 + h) * MM + c];
    s1 += w * W1[(66 + h) * MM + c];
  }
  c0[c] = s0; c1[c] = s1;
  if (c < LL) {
    float sm = 0.f, st = 0.f;
    for (int h = 0; h < LL; ++h) sm += We[h] * Wtm[(64 + h) * LL + c];
    for (int m = 0; m < LL; ++m) st += Wtu[c * LL + m] * Wterm[m];
    cm[c] = sm; wtt[c] = st;
  }
  if (c == 0) {
    float sp = 0.f;
    for (int h = 0; h < LL; ++h) sp += We[h] * Wpred[64 + h];
    cp[0] = sp;
  }
}

// ---------------------------------------------------------------------------
// Kernel 2: heavy pairwise message GEMM + masked max + update + decoder.
// One block per (b,i): 256 threads = 8 waves, wave w handles j-strips w, w+8.
//
// WMMA fragment-native LDS layout (V_WMMA_F32_16X16X4_F32, wave32):
//   lane l needs K pair {k0+2*(l>>4), k0+2*(l>>4)+1} for row/col (l&15).
//   group g = kk*2 + sel  (sel = l>>4) stores that pair contiguously:
//     A strip : sRf[g*32  + row*2 + pos]   (stride 32  -> halves on banks 0-31/32-63)
//     W2      : sW2f[g*224 + col*2 + pos]  (224%64==32 -> disjoint bank halves)
//   => inner loop is a single aligned ds_load_b64 per operand, conflict-free.
// ---------------------------------------------------------------------------
__global__ void ne3_main(const float* __restrict__ xs, const float* __restrict__ hs,
                         const float* __restrict__ ef, const int* __restrict__ adj,
                         const float* __restrict__ A0, const float* __restrict__ B0,
                         const float* __restrict__ A1, const float* __restrict__ B1,
                         const float* __restrict__ C0, const float* __restrict__ C1,
                         const float* __restrict__ W2, const float* __restrict__ Wpu,
                         const float* __restrict__ Wdec,
                         float* __restrict__ out_newx, float* __restrict__ out_cath,
                         float* __restrict__ newh) {
  extern __shared__ float sm[];
  float* sW2f = sm;                     // 48*224 = 10752
  float* sRf  = sW2f + 48 * W2G;        // 8*1536 = 12288
  float* sPt  = sRf + 8 * (48 * RFG);   // 768
  float* sA0  = sPt + 8 * 96;           // 96
  float* sA1  = sA0 + 96;               // 96
  float* sC0  = sA1 + 96;               // 96
  float* sC1  = sC0 + 96;               // 96
  float* sE   = sC1 + 96;               // 256
  float* sMk  = sE + 256;               // 256
  float* sAgg = sMk + 256;              // 96
  float* sNH  = sAgg + 96;              // 32
  float* sZ   = sNH + 32;               // 34

  const int tid = threadIdx.x;
  const int bi = blockIdx.x;        // b*N + i
  const int b = bi >> 8;
  const int i = bi & (NN - 1);
  const int jb = b << 8;            // base row index for j nodes

  // stage W2 into fragment layout
  for (int idx = tid; idx < MM * MM; idx += 256) {
    const int r = idx / MM, c = idx - r * MM;   // r = K index
    const int kk = r >> 2, kr = r & 3;
    const int sel = kr >> 1, pos = kr & 1;
    sW2f[(kk * 2 + sel) * W2G + c * 2 + pos] = W2[idx];
  }
  for (int c = tid; c < MM; c += 256) {
    sA0[c] = A0[bi * MM + c];
    sA1[c] = A1[bi * MM + c];
    sC0[c] = C0[c];
    sC1[c] = C1[c];
  }
  {
    const int j = tid;  // 256 threads == N
    sE[j] = ef[(size_t)bi * NN + j];
    const int a2 = adj[(size_t)bi * NN + j] + (j == i ? 1 : 0);
    sMk[j] = (a2 > 0) ? 1.f : 0.f;
  }
  if (tid < 33) sZ[tid] = (tid == 0) ? xs[bi] : hs[bi * LL + tid - 1];
  __syncthreads();

  const int wid = tid >> 5;
  const int lane = tid & 31;
  const int mrow = lane & 15;           // M / N index within tile
  const int selA = lane >> 4;           // which K pair of the 4-chunk
  const int rowoff = selA << 3;         // C/D: lanes 16-31 hold rows M+8
  float* mysRf = sRf + wid * (48 * RFG);

  float wmax[6];
#pragma unroll
  for (int t = 0; t < 6; ++t) wmax[t] = neg_inf();

  for (int s = wid; s < 16; s += 8) {
    const int j0 = s << 4;
    // build relu pre-activation strip directly in fragment layout
    for (int idx = lane; idx < 16 * MM; idx += 32) {
      const int jr = idx / MM, c = idx - jr * MM;   // c = K index
      const int j = j0 + jr;
      const float v = sA1[c] + B1[(size_t)(jb + j) * MM + c] + sE[j] * sC1[c];
      const int kk = c >> 2, kr = c & 3;
      const int sel = kr >> 1, pos = kr & 1;
      mysRf[(kk * 2 + sel) * RFG + jr * 2 + pos] = v > 0.f ? v : 0.f;
    }
    // wave-local LDS RAW fence before matrix reads
    asm volatile("s_wait_dscnt 0" ::: "memory");

    v8f acc[6];
#pragma unroll
    for (int t = 0; t < 6; ++t)
#pragma unroll
      for (int q = 0; q < 8; ++q) acc[t][q] = 0.f;

    for (int kk = 0; kk < NKK; ++kk) {
      const int g = kk * 2 + selA;
      const v2f a = *(const v2f*)(mysRf + g * RFG + mrow * 2);
      const float* wb = sW2f + g * W2G + mrow * 2;
#pragma unroll
      for (int t = 0; t < 6; ++t) {
        const v2f bf = *(const v2f*)(wb + t * 32);
        acc[t] = __builtin_amdgcn_wmma_f32_16x16x4_f32(
            false, a, false, bf, (short)0, acc[t], false, false);
      }
    }

    // epilogue: add linear branch, mask, running max over this strip's rows
#pragma unroll
    for (int t = 0; t < 6; ++t) {
      const int col = t * 16 + mrow;
      const float a0c = sA0[col];
      const float c0c = sC0[col];
      float lm = neg_inf();
#pragma unroll
      for (int r = 0; r < 8; ++r) {
        const int j = j0 + r + rowoff;
        float val = acc[t][r] + a0c + B0[(size_t)(jb + j) * MM + col] + sE[j] * c0c;
        val = (sMk[j] > 0.f) ? val : neg_inf();
        lm = fmaxf(lm, val);
      }
      lm = fmaxf(lm, __shfl_xor(lm, 16, 32));  // merge rows M and M+8 halves
      wmax[t] = fmaxf(wmax[t], lm);
    }
  }

  if (lane < 16) {
#pragma unroll
    for (int t = 0; t < 6; ++t) sPt[wid * 96 + t * 16 + lane] = wmax[t];
  }
  __syncthreads();

  if (tid < MM) {
    float mx = neg_inf();
#pragma unroll
    for (int w = 0; w < 8; ++w) mx = fmaxf(mx, sPt[w * 96 + tid]);
    sAgg[tid] = mx;
  }
  __syncthreads();

  if (tid < LL) {
    float acc = 0.f;
    for (int k = 0; k < MM; ++k) acc += sAgg[k] * Wpu[k * LL + tid];
    newh[bi * LL + tid] = acc;
    out_cath[bi * LL + tid] = acc;
    sNH[tid] = acc;
  }
  __syncthreads();

  if (tid == 0) {
    float acc = 0.f;
    for (int k = 0; k < 33; ++k) acc += sZ[k] * Wdec[k];
    for (int k = 0; k < LL; ++k) acc += sNH[k] * Wdec[33 + k];
    out_newx[bi] = acc;
  }
}

// ---------------------------------------------------------------------------
// Kernel 3a: per-node projections of new_h for the pred/termination heads.
// Hm = h@Wtm[0:32], Gm = h@Wtm[32:64], S = h@Wpred[0:32], T = h@Wpred[32:64]
// grid = B*N blocks, 32 threads
// ---------------------------------------------------------------------------
__global__ void ne3_post_proj(const float* __restrict__ newh,
                              const float* __restrict__ Wtm, const float* __restrict__ Wpred,
                              float* __restrict__ Hm, float* __restrict__ Gm,
                              float* __restrict__ S, float* __restrict__ T) {
  __shared__ float h[LL];
  const int bi = blockIdx.x;
  const int tid = threadIdx.x;
  h[tid] = newh[bi * LL + tid];
  __syncthreads();
  float hm = 0.f, gm = 0.f;
  for (int k = 0; k < LL; ++k) {
    hm += h[k] * Wtm[k * LL + tid];
    gm += h[k] * Wtm[(LL + k) * LL + tid];
  }
  Hm[bi * LL + tid] = hm;
  Gm[bi * LL + tid] = gm;
  if (tid == 0) { float s = 0.f; for (int k = 0; k < LL; ++k) s += h[k] * Wpred[k]; S[bi] = s; }
  if (tid == 1) { float t = 0.f; for (int k = 0; k < LL; ++k) t += h[k] * Wpred[LL + k]; T[bi] = t; }
}

// ---------------------------------------------------------------------------
// Kernel 3b: pairwise outputs via the rank-1 decomposition.
//   p[b,i,j] = mask ? S_i + T_j + e_ij*cp : -inf
//   maxq[c]  = max_{masked j} relu(Hm_i[c] + Gm_j[c] + e_ij*cm[c])
//   tauc[b,i]= maxq . wtt      (wtt = Wtu@Wterm)
// grid = B*N blocks, 256 threads (8 j-groups x 32 cols)
// ---------------------------------------------------------------------------
__global__ void ne3_pairs(const float* __restrict__ ef, const int* __restrict__ adj,
                          const float* __restrict__ Hm, const float* __restrict__ Gm,
                          const float* __restrict__ S, const float* __restrict__ T,
                          const float* __restrict__ cm, const float* __restrict__ wtt,
                          const float* __restrict__ cp,
                          float* __restrict__ out_p, float* __restrict__ tauc) {
  __shared__ float sHm[LL], sCm[LL], sWtt[LL], sMaxq[LL];
  __shared__ float sEr[NN], sMk[NN];
  __shared__ float sPart[8 * LL];
  __shared__ float sSi[1];
  const int bi = blockIdx.x;
  const int b = bi >> 8;
  const int i = bi & (NN - 1);
  const int jb = b << 8;
  const int tid = threadIdx.x;

  if (tid < LL) { sHm[tid] = Hm[bi * LL + tid]; sCm[tid] = cm[tid]; sWtt[tid] = wtt[tid]; }
  if (tid == 0) sSi[0] = S[bi];
  {
    const int j = tid;
    sEr[j] = ef[(size_t)bi * NN + j];
    const int a2 = adj[(size_t)bi * NN + j] + (j == i ? 1 : 0);
    sMk[j] = (a2 > 0) ? 1.f : 0.f;
  }
  __syncthreads();

  // p row
  {
    const int j = tid;
    const float pv = (sMk[j] > 0.f) ? (sSi[0] + T[jb + j] + sEr[j] * cp[0]) : neg_inf();
    out_p[(size_t)bi * NN + j] = pv;
  }

  // termination message masked max
  const int col = tid & (LL - 1);
  const int jg = tid >> 5;  // 8 groups of 32 j's
  float mx = neg_inf();
  const float hmc = sHm[col], cmc = sCm[col];
  for (int jj = 0; jj < 32; ++jj) {
    const int j = (jg << 5) + jj;
    if (sMk[j] > 0.f) {
      float v = hmc + Gm[(size_t)(jb + j) * LL + col] + sEr[j] * cmc;
      v = v > 0.f ? v : 0.f;
      mx = fmaxf(mx, v);
    }
  }
  sPart[jg * LL + col] = mx;
  __syncthreads();
  if (tid < LL) {
    float m2 = neg_inf();
#pragma unroll
    for (int g = 0; g < 8; ++g) m2 = fmaxf(m2, sPart[g * LL + tid]);
    sMaxq[tid] = m2;
  }
  __syncthreads();
  if (tid == 0) {
    float acc = 0.f;
    for (int c = 0; c < LL; ++c) acc += sMaxq[c] * sWtt[c];
    tauc[bi] = acc;
  }
}

// ---------------------------------------------------------------------------
// Kernel 3c: tau[b] = mean_i tauc[b,i] (deterministic tree reduction)
// grid = B blocks, 256 threads
// ---------------------------------------------------------------------------
__global__ void ne3_tau(const float* __restrict__ tauc, float* __restrict__ out_tau) {
  __shared__ float red[NN];
  const int b = blockIdx.x;
  const int tid = threadIdx.x;
  red[tid] = tauc[b * NN + tid];
  __syncthreads();
  for (int st = 128; st > 0; st >>= 1) {
    if (tid < st) red[tid] += red[tid + st];
    __syncthreads();
  }
  if (tid == 0) out_tau[b] = red[0] * (1.f / (float)NN);
}

// ---------------------------------------------------------------------------
extern "C" void kernel_launch(void* const* d_in, const int* in_sizes, int n_in,
                              void* d_out, int out_size, void* d_ws, size_t ws_size,
                              hipStream_t stream) {
  (void)in_sizes; (void)n_in; (void)out_size; (void)ws_size;
  const float* xs    = (const float*)d_in[0];
  const float* hs    = (const float*)d_in[1];
  const float* ef    = (const float*)d_in[2];
  const int*   adj   = (const int*)d_in[3];
  const float* We    = (const float*)d_in[4];
  const float* Wlin  = (const float*)d_in[5];
  const float* W1    = (const float*)d_in[6];
  const float* W2    = (const float*)d_in[7];
  const float* Wpu   = (const float*)d_in[8];
  const float* Wdec  = (const float*)d_in[9];
  const float* Wpred = (const float*)d_in[10];
  const float* Wtm   = (const float*)d_in[11];
  const float* Wtu   = (const float*)d_in[12];
  const float* Wterm = (const float*)d_in[13];

  // output layout: new_x [B*N] | p [B*N*N] | tau [B] | cat_h [B*N*L]
  float* out = (float*)d_out;
  float* out_newx = out;
  float* out_p    = out + BB * NN;
  float* out_tau  = out_p + (size_t)BB * NN * NN;
  float* out_cath = out_tau + BB;

  // workspace carve (floats)
  float* ws = (float*)d_ws;
  const size_t RN = (size_t)BB * NN;  // 2048 node rows
  float* A0   = ws;                 // RN*96
  float* B0   = A0 + RN * MM;
  float* A1   = B0 + RN * MM;
  float* B1   = A1 + RN * MM;
  float* newh = B1 + RN * MM;       // RN*32
  float* Hm   = newh + RN * LL;
  float* Gm   = Hm + RN * LL;
  float* S    = Gm + RN * LL;       // RN
  float* T    = S + RN;
  float* c0   = T + RN;             // 96
  float* c1   = c0 + MM;
  float* cm   = c1 + MM;            // 32
  float* wtt  = cm + LL;
  float* cp   = wtt + LL;           // 1 (+pad)
  float* tauc = cp + 8;             // RN

  ne3_pre<<<dim3(BB * NN), dim3(96), 0, stream>>>(xs, hs, Wlin, W1, A0, B0, A1, B1);
  ne3_consts<<<dim3(1), dim3(96), 0, stream>>>(We, Wlin, W1, Wtm, Wtu, Wterm, Wpred,
                                               c0, c1, cm, wtt, cp);

  const size_t smem = (size_t)(48 * W2G + 8 * 48 * RFG + 8 * 96 + 4 * 96 + 2 * 256 +
                               96 + 32 + 34) * sizeof(float);
  ne3_main<<<dim3(BB * NN), dim3(256), smem, stream>>>(
      xs, hs, ef, adj, A0, B0, A1, B1, c0, c1, W2, Wpu, Wdec,
      out_newx, out_cath, newh);

  ne3_post_proj<<<dim3(BB * NN), dim3(32), 0, stream>>>(newh, Wtm, Wpred, Hm, Gm, S, T);
  ne3_pairs<<<dim3(BB * NN), dim3(256), 0, stream>>>(ef, adj, Hm, Gm, S, T, cm, wtt, cp,
                                                     out_p, tauc);
  ne3_tau<<<dim3(BB), dim3(256), 0, stream>>>(tauc, out_tau);
}